// Graph_ReLu_W_WithPrior_11940009082915
// MI455X (gfx1250) — compile-verified
//
#include <hip/hip_runtime.h>
#include <stdint.h>

// Graph_ReLu_W_WithPrior: out[i,j] = relu(A[i,j]) kept only at the row-wise
// top-K (K=32) positions of relu(A), else 0. N = 8192, fp32.
//
// Bandwidth-bound: 256MB read + 256MB write -> ~22us floor at 23.3 TB/s.
// CDNA5 paths used: global_load_async_to_lds_b128 (ASYNCcnt) with TH_LOAD_NT,
// split barriers, LDS atomics, wave32 shuffle suffix-scan, non-temporal b128
// stores. WMMA intentionally unused: the op has no matmul content.

#define NN    8192
#define KK    32
#define TPB   256
#define NWAVE (TPB / 32)
#define VEC_ITERS (NN / (TPB * 4))   // 8

typedef __attribute__((ext_vector_type(4))) float v4f;

__global__ __launch_bounds__(TPB)
void topk_row_mask_kernel(const float* __restrict__ A, float* __restrict__ Out) {
    const int row  = blockIdx.x;
    const int t    = threadIdx.x;
    const int wave = t >> 5;
    const int lane = t & 31;

    __shared__ __align__(16) float srow[NN];              // 32 KB row stage
    __shared__ unsigned int hist8[NWAVE * 256];           // per-wave histograms
    __shared__ unsigned int sPrefix, sQuota, sEq;

    // ---------------- Phase 1: async copy row (32 KB) global -> LDS --------
    // Instruction offset advances BOTH the global and LDS address (ISA
    // CDNA5 ISA), so one base pair covers the whole row.
    {
        const float* rowp = A + (size_t)row * NN;
        uint64_t gaddr = (uint64_t)(uintptr_t)rowp + (uint64_t)(t * 16);
        uint32_t laddr = (uint32_t)(uintptr_t)srow + (uint32_t)(t * 16);
        asm volatile(
            "global_load_async_to_lds_b128 %0, %1, off th:TH_LOAD_NT\n\t"
            "global_load_async_to_lds_b128 %0, %1, off offset:4096 th:TH_LOAD_NT\n\t"
            "global_load_async_to_lds_b128 %0, %1, off offset:8192 th:TH_LOAD_NT\n\t"
            "global_load_async_to_lds_b128 %0, %1, off offset:12288 th:TH_LOAD_NT\n\t"
            "global_load_async_to_lds_b128 %0, %1, off offset:16384 th:TH_LOAD_NT\n\t"
            "global_load_async_to_lds_b128 %0, %1, off offset:20480 th:TH_LOAD_NT\n\t"
            "global_load_async_to_lds_b128 %0, %1, off offset:24576 th:TH_LOAD_NT\n\t"
            "global_load_async_to_lds_b128 %0, %1, off offset:28672 th:TH_LOAD_NT\n\t"
            "s_wait_asynccnt 0x0\n\t"
            :
            : "v"(laddr), "v"(gaddr)
            : "memory");
    }
    if (t == 0) { sPrefix = 0u; sQuota = KK; sEq = 0u; }
    __syncthreads();

    // ---------------- Phase 2: 4-pass radix select (find 32nd largest) -----
    // key(v) = bits(v) if v > 0 else 0 (positive IEEE floats order as uint32).
    for (int shift = 24; shift >= 0; shift -= 8) {
        const unsigned int prefix = sPrefix;   // stable: barrier since last write
        const unsigned int q      = sQuota;

        #pragma unroll
        for (int w = 0; w < NWAVE; ++w) hist8[w * 256 + t] = 0u;
        __syncthreads();

        const unsigned int hiKeep = (shift == 24) ? 0u : (prefix >> (shift + 8));
        unsigned int* myHist = &hist8[wave * 256];
        unsigned int zc = 0u;                   // locally aggregated bin-0 count

        #pragma unroll
        for (int j = 0; j < VEC_ITERS; ++j) {
            int i4 = j * TPB + t;
            v4f v = *(const v4f*)&srow[i4 * 4];
            #pragma unroll
            for (int c = 0; c < 4; ++c) {
                float f = v[c];
                unsigned int key = (f > 0.0f) ? __float_as_uint(f) : 0u;
                bool in = (shift == 24) || ((key >> (shift + 8)) == hiKeep);
                if (in) {
                    if (key == 0u) zc++;        // avoid 4096-way bin-0 pileup
                    else atomicAdd(&myHist[(key >> shift) & 0xFFu], 1u);
                }
            }
        }
        if (zc) atomicAdd(&myHist[0], zc);
        __syncthreads();

        // Wave 0: fold per-wave histograms, wave-wide suffix-scan, pick bin.
        if (wave == 0) {
            unsigned int h[8];
            unsigned int lsum = 0u;
            #pragma unroll
            for (int k = 0; k < 8; ++k) {
                int b = lane * 8 + k;
                unsigned int s = 0u;
                #pragma unroll
                for (int w = 0; w < NWAVE; ++w) s += hist8[w * 256 + b];
                h[k] = s;
                lsum += s;
            }
            // Inclusive suffix-sum of lane sums: suf[l] = sum over lanes >= l.
            unsigned int suf = lsum;
            #pragma unroll
            for (int d = 1; d < 32; d <<= 1) {
                unsigned int up = __shfl_down(suf, d, 32);
                if (lane + d < 32) suf += up;
            }
            const unsigned int above = suf - lsum;  // lanes strictly above
            // SufBin[lane*8+k] = above + sum_{k'>=k} h[k'], nonincreasing in b.
            unsigned int sufbin[9];
            sufbin[8] = above;
            unsigned int run = above;
            #pragma unroll
            for (int k = 7; k >= 0; --k) { run += h[k]; sufbin[k] = run; }
            #pragma unroll
            for (int k = 0; k < 8; ++k) {
                if (sufbin[k] >= q && sufbin[k + 1] < q) {   // unique winner
                    sPrefix = prefix | ((unsigned int)(lane * 8 + k) << shift);
                    sQuota  = q - sufbin[k + 1];
                }
            }
        }
        __syncthreads();
    }

    const unsigned int T       = sPrefix;
    const unsigned int quotaEq = sQuota;

    // ---------------- Phase 3: masked non-temporal streaming write ----------
    float* orow = Out + (size_t)row * NN;
    #pragma unroll
    for (int j = 0; j < VEC_ITERS; ++j) {
        int i4 = j * TPB + t;
        v4f v = *(const v4f*)&srow[i4 * 4];
        v4f o;
        #pragma unroll
        for (int c = 0; c < 4; ++c) {
            float f  = v[c];
            float rv = (f > 0.0f) ? f : 0.0f;
            unsigned int key = (f > 0.0f) ? __float_as_uint(f) : 0u;
            float oc = 0.0f;
            if (key > T) {
                oc = rv;
            } else if (key == T && T != 0u) {   // T==0 ties emit 0.0 anyway
                unsigned int old = atomicAdd(&sEq, 1u);
                if (old < quotaEq) oc = rv;
            }
            o[c] = oc;
        }
        __builtin_nontemporal_store(o, (v4f*)(orow + i4 * 4));
    }
}

extern "C" void kernel_launch(void* const* d_in, const int* in_sizes, int n_in,
                              void* d_out, int out_size, void* d_ws, size_t ws_size,
                              hipStream_t stream) {
    const float* A = (const float*)d_in[0];
    // d_in[1] (idx) is unused by the reference forward pass.
    float* Out = (float*)d_out;
    (void)in_sizes; (void)n_in; (void)out_size; (void)d_ws; (void)ws_size;

    dim3 grid(NN);
    dim3 block(TPB);
    topk_row_mask_kernel<<<grid, block, 0, stream>>>(A, Out);
}